// MeanAggregator_21887153340603
// MI455X (gfx1250) — compile-verified
//
#include <hip/hip_runtime.h>

// MeanAggregator SpMM, fixed degree:
//   out[i,:] = sum_j vals[6i+j] * x[cols[6i+j],:] / sum_j vals[6i+j]
// Memory-bound gather. One wave32 per node; lane l handles floats [4l..4l+3]
// of the D=128 feature row (global_load_b128 per lane -> 512B per neighbor row).
// x (51MB) is kept L2-resident (192MB L2); output is streamed with NT stores.

#define DEG 6
#define DIM 128
#define V4  (DIM / 4)   // 32 float4 per feature row == one per lane

typedef float v4f __attribute__((ext_vector_type(4)));

__global__ __launch_bounds__(256, 8)
void mean_agg_spmm_kernel(const float* __restrict__ x,
                          const int*   __restrict__ cols,
                          const float* __restrict__ vals,
                          float*       __restrict__ out,
                          int n_nodes) {
  const int lane = threadIdx.x & 31;
  // Node id is wave-uniform: force it into an SGPR so cols/vals loads scalarize.
  const int node = __builtin_amdgcn_readfirstlane(
      (int)blockIdx.x * ((int)blockDim.x >> 5) + ((int)threadIdx.x >> 5));
  if (node >= n_nodes) return;

  const long ebase = (long)node * DEG;

  // Edge indices + weights (wave-uniform addresses -> SMEM path).
  int   c[DEG];
  float w[DEG];
#pragma unroll
  for (int j = 0; j < DEG; ++j) {
    c[j] = cols[ebase + j];
    w[j] = vals[ebase + j];
  }
  const float wsum = ((w[0] + w[1]) + (w[2] + w[3])) + (w[4] + w[5]);

  // Issue all 6 gathers first (6 independent global_load_b128 in flight),
  // then accumulate. x stays in L2 (51MB < 192MB).
  const v4f* __restrict__ xv = (const v4f*)x;
  v4f r[DEG];
#pragma unroll
  for (int j = 0; j < DEG; ++j) {
    r[j] = xv[(long)c[j] * V4 + lane];
  }

  v4f acc = (v4f)0.0f;
#pragma unroll
  for (int j = 0; j < DEG; ++j) {
    acc += w[j] * r[j];
  }
  acc *= (1.0f / wsum);

  // Write-once output: non-temporal b128 store so it doesn't evict x from L2.
  v4f* outv = (v4f*)out;
  __builtin_nontemporal_store(acc, &outv[(long)node * V4 + lane]);
}

extern "C" void kernel_launch(void* const* d_in, const int* in_sizes, int n_in,
                              void* d_out, int out_size, void* d_ws, size_t ws_size,
                              hipStream_t stream) {
  // setup_inputs order: x [N*D] f32, rows [E] i32 (implicit: rows[e]=e/DEG,
  // sorted & contiguous -> unused), cols [E] i32, vals [E] f32.
  const float* x    = (const float*)d_in[0];
  const int*   cols = (const int*)  d_in[2];
  const float* vals = (const float*)d_in[3];
  float*       out  = (float*)d_out;

  const int n_nodes = in_sizes[0] / DIM;          // N = 100000
  const int waves_per_block = 256 / 32;           // 8 nodes per block
  const int grid = (n_nodes + waves_per_block - 1) / waves_per_block;

  mean_agg_spmm_kernel<<<grid, 256, 0, stream>>>(x, cols, vals, out, n_nodes);
}